// GCN_79431125172489
// MI455X (gfx1250) — compile-verified
//
#include <hip/hip_runtime.h>
#include <hip/hip_bf16.h>

typedef __attribute__((ext_vector_type(2))) float v2f;
typedef __attribute__((ext_vector_type(8))) float v8f;

static __host__ inline int cdiv(int a, int b) { return (a + b - 1) / b; }

// ---------------------------------------------------------------------------
// Degree / normalization
// ---------------------------------------------------------------------------
__global__ void k_init_deg(float* __restrict__ dinv, int n) {
    int i = blockIdx.x * blockDim.x + threadIdx.x;
    if (i < n) dinv[i] = 1.0f;  // self-loop contributes 1 to every node's degree
}

__global__ void k_count_deg(const int* __restrict__ ei, float* __restrict__ dinv, int E) {
    int e = blockIdx.x * blockDim.x + threadIdx.x;
    if (e < E) atomicAdd(&dinv[ei[E + e]], 1.0f);  // dst row of edge_index
}

__global__ void k_finalize_dinv(float* __restrict__ dinv, int n) {
    int i = blockIdx.x * blockDim.x + threadIdx.x;
    if (i < n) dinv[i] = rsqrtf(dinv[i]);  // deg >= 1 always (self loops)
}

// ---------------------------------------------------------------------------
// Layer 1 dense transform: xw[N,16] = x[N,3] @ W1[3,16]  (one WMMA per 16 nodes)
// A = 16x4 tile of x (K=3, 4th K zero), B = W1 padded to 4x16, D = 16x16 f32.
// Branchless operand setup: WMMA needs EXEC all-ones, so avoid divergence.
// ---------------------------------------------------------------------------
__global__ void k_xform1_wmma(const float* __restrict__ x, const float* __restrict__ W1,
                              float* __restrict__ xw, int n) {
    int lane = threadIdx.x & 31;
    int wave = (blockIdx.x * blockDim.x + threadIdx.x) >> 5;
    int base = wave * 16;            // first node (M row) of this wave's tile
    int half = lane >> 4;            // 0: K={0,1}; 1: K={2,3(pad)}
    int l    = lane & 15;
    int node = base + l;
    int nc   = node < n ? node : (n - 1);  // clamp loads; tail stores are guarded

    float x0 = x[nc * 3 + 0];
    float x1 = x[nc * 3 + 1];
    float x2 = x[nc * 3 + 2];
    float w0 = W1[0 * 16 + l];
    float w1 = W1[1 * 16 + l];
    float w2 = W1[2 * 16 + l];

    v2f a, b;
    a.x = half ? x2 : x0;  a.y = half ? 0.0f : x1;
    b.x = half ? w2 : w0;  b.y = half ? 0.0f : w1;

    v8f c = {};
    c = __builtin_amdgcn_wmma_f32_16x16x4_f32(false, a, false, b, (short)0, c, false, false);

    // D layout: VGPR r -> node row (base + r + 8*half), feature = l
    int row0 = base + 8 * half;
    if (base + 16 <= n) {            // wave-uniform fast path (N % 16 == 0 case)
#pragma unroll
        for (int r = 0; r < 8; ++r) xw[(row0 + r) * 16 + l] = c[r];
    } else {
#pragma unroll
        for (int r = 0; r < 8; ++r)
            if (row0 + r < n) xw[(row0 + r) * 16 + l] = c[r];
    }
}

// ---------------------------------------------------------------------------
// Self-loop seed for layer-1 accumulator: h2[i,:] = xw[i,:] * dinv[i]^2 (float4)
// ---------------------------------------------------------------------------
__global__ void k_selfloop1(const float4* __restrict__ xw4, const float* __restrict__ dinv,
                            float4* __restrict__ h24, int n) {
    int i = blockIdx.x * blockDim.x + threadIdx.x;   // over n*4 float4 chunks
    if (i < n * 4) {
        float di = dinv[i >> 2];
        float s  = di * di;
        float4 v = xw4[i];
        v.x *= s; v.y *= s; v.z *= s; v.w *= s;
        h24[i] = v;
    }
}

// ---------------------------------------------------------------------------
// Layer 1 edge scatter: 16 lanes per edge, one feature per lane.
// A wave's atomic instruction covers 2 edges -> 2 destination rows (2 CLs).
// ---------------------------------------------------------------------------
__global__ void k_scatter1(const int* __restrict__ ei, const float* __restrict__ xw,
                           const float* __restrict__ dinv, float* __restrict__ h2, int E) {
    int t = blockIdx.x * blockDim.x + threadIdx.x;
    if (t < E * 16) {
        int e = t >> 4, k = t & 15;
        int s = ei[e], d = ei[E + e];
        float nm = dinv[s] * dinv[d];
        atomicAdd(&h2[d * 16 + k], xw[s * 16 + k] * nm);
    }
}

__global__ void k_relu_bias(float4* __restrict__ h24, const float4* __restrict__ b14, int n) {
    int i = blockIdx.x * blockDim.x + threadIdx.x;   // over n*4 float4 chunks
    if (i < n * 4) {
        float4 v = h24[i];
        float4 b = b14[i & 3];
        v.x = fmaxf(v.x + b.x, 0.0f);
        v.y = fmaxf(v.y + b.y, 0.0f);
        v.z = fmaxf(v.z + b.z, 0.0f);
        v.w = fmaxf(v.w + b.w, 0.0f);
        h24[i] = v;
    }
}

// ---------------------------------------------------------------------------
// Layer 2 dense transform: hw[N,8pad] = h[N,16] @ W2[16,7]
// Four chained K=4 WMMAs accumulate the full K=16 product.
// A rows loaded as float4 (b128); W2 columns masked branchlessly.
// ---------------------------------------------------------------------------
__global__ void k_xform2_wmma(const float4* __restrict__ h4, const float* __restrict__ W2,
                              float* __restrict__ hw, int n) {
    int lane = threadIdx.x & 31;
    int wave = (blockIdx.x * blockDim.x + threadIdx.x) >> 5;
    int base = wave * 16;
    int half = lane >> 4;
    int l    = lane & 15;
    int node = base + l;
    int nc   = node < n ? node : (n - 1);

    int   li = l < 7 ? l : 6;           // clamped column for unconditional load
    float lm = l < 7 ? 1.0f : 0.0f;     // mask for pad columns

    v8f c = {};
#pragma unroll
    for (int kk = 0; kk < 4; ++kk) {
        float4 hv = h4[nc * 4 + kk];    // h[nc, kk*4 .. kk*4+3]
        int    k0 = kk * 4 + half * 2;  // this lane's two K indices: k0, k0+1
        v2f a, b;
        a.x = half ? hv.z : hv.x;
        a.y = half ? hv.w : hv.y;
        b.x = W2[(k0)     * 7 + li] * lm;   // W2 is [16,7] row-major
        b.y = W2[(k0 + 1) * 7 + li] * lm;
        c = __builtin_amdgcn_wmma_f32_16x16x4_f32(false, a, false, b, (short)0, c, false, false);
    }

    int row0 = base + 8 * half;
    if (base + 16 <= n) {
        if (l < 7) {
#pragma unroll
            for (int r = 0; r < 8; ++r) hw[(row0 + r) * 8 + l] = c[r];
        }
    } else {
#pragma unroll
        for (int r = 0; r < 8; ++r)
            if (row0 + r < n && l < 7) hw[(row0 + r) * 8 + l] = c[r];
    }
}

// ---------------------------------------------------------------------------
// Seed output with bias + self-loop term: out[i,k] = b2[k] + hw[i,k]*dinv[i]^2
// ---------------------------------------------------------------------------
__global__ void k_init_out(const float* __restrict__ hw, const float* __restrict__ dinv,
                           const float* __restrict__ b2, float* __restrict__ out, int n) {
    int i = blockIdx.x * blockDim.x + threadIdx.x;
    if (i < n * 7) {
        int node = i / 7, k = i - node * 7;
        float di = dinv[node];
        out[i] = b2[k] + hw[node * 8 + k] * di * di;
    }
}

// ---------------------------------------------------------------------------
// Layer 2 edge scatter: 8 lanes per edge (7 active features).
// ---------------------------------------------------------------------------
__global__ void k_scatter2(const int* __restrict__ ei, const float* __restrict__ hw,
                           const float* __restrict__ dinv, float* __restrict__ out, int E) {
    int t = blockIdx.x * blockDim.x + threadIdx.x;
    if (t < E * 8) {
        int e = t >> 3, k = t & 7;
        if (k < 7) {
            int s = ei[e], d = ei[E + e];
            float nm = dinv[s] * dinv[d];
            atomicAdd(&out[d * 7 + k], hw[s * 8 + k] * nm);
        }
    }
}

// ---------------------------------------------------------------------------
// Launch
// ---------------------------------------------------------------------------
extern "C" void kernel_launch(void* const* d_in, const int* in_sizes, int n_in,
                              void* d_out, int out_size, void* d_ws, size_t ws_size,
                              hipStream_t stream) {
    const float* x  = (const float*)d_in[0];
    const int*   ei = (const int*)d_in[1];
    const float* W1 = (const float*)d_in[2];
    const float* b1 = (const float*)d_in[3];
    const float* W2 = (const float*)d_in[4];
    const float* b2 = (const float*)d_in[5];
    float* out = (float*)d_out;

    const int N = in_sizes[0] / 3;   // 100000
    const int E = in_sizes[1] / 2;   // 3200000

    // Workspace layout (floats): dinv[N] | buf1[16N] (xw, reused as hw) | buf2[16N] (h2/h)
    float* dinv = (float*)d_ws;
    float* buf1 = dinv + N;
    float* buf2 = buf1 + (size_t)N * 16;

    const int B = 256;

    // 1) degrees -> dinv
    k_init_deg<<<cdiv(N, B), B, 0, stream>>>(dinv, N);
    k_count_deg<<<cdiv(E, B), B, 0, stream>>>(ei, dinv, E);
    k_finalize_dinv<<<cdiv(N, B), B, 0, stream>>>(dinv, N);

    // 2) layer 1: xw = x@W1 (WMMA), self-loop seed, scatter, bias+relu
    int waves1 = cdiv(N, 16);
    k_xform1_wmma<<<cdiv(waves1 * 32, B), B, 0, stream>>>(x, W1, buf1, N);
    k_selfloop1<<<cdiv(N * 4, B), B, 0, stream>>>((const float4*)buf1, dinv, (float4*)buf2, N);
    k_scatter1<<<cdiv(E * 16, B), B, 0, stream>>>(ei, buf1, dinv, buf2, E);
    k_relu_bias<<<cdiv(N * 4, B), B, 0, stream>>>((float4*)buf2, (const float4*)b1, N);

    // 3) layer 2: hw = h@W2 (4 chained WMMAs), seed out, scatter
    k_xform2_wmma<<<cdiv(waves1 * 32, B), B, 0, stream>>>((const float4*)buf2, W2, buf1, N);
    k_init_out<<<cdiv(N * 7, B), B, 0, stream>>>(buf1, dinv, b2, out, N);
    k_scatter2<<<cdiv(E * 8, B), B, 0, stream>>>(ei, buf1, dinv, out, E);
}